// Spiking1DeALIFLayer_8246337208354
// MI455X (gfx1250) — compile-verified
//
#include <hip/hip_runtime.h>
#include <hip/hip_bf16.h>

// eALIF spiking scan, [B,C,T] float32, sequential over T per (b,c) row.
// Memory-bound: 268 MB total traffic -> ~11.5 us floor at 23.3 TB/s.
// Strategy: stage [R rows x TC timesteps] tiles through LDS with async
// global->LDS b128 loads (double buffered, ASYNCcnt), scan rows from LDS,
// overwrite with spikes in place, store tiles back coalesced (non-temporal).

#define EALIF_BETA  0.65f
#define EALIF_GAMMA 0.75f
#define EALIF_RHO   0.75f

constexpr int R      = 128;     // rows per block == threads per block (4 waves)
constexpr int TC     = 32;      // timesteps per LDS-staged chunk
constexpr int F4ROW  = TC / 4;  // 8 float4 per row-chunk
constexpr int LPT    = F4ROW;   // async-load instructions per thread per chunk
constexpr int STRIDE = TC + 4;  // 36 floats = 144B pitch: 16B aligned, avoids 32-way LDS conflicts

typedef float f4  __attribute__((ext_vector_type(4)));
typedef int   v4i __attribute__((ext_vector_type(4)));

#if __has_builtin(__builtin_amdgcn_global_load_async_to_lds_b128)
#define HAVE_ASYNC_LDS 1
#else
#define HAVE_ASYNC_LDS 0
#endif

#if __has_builtin(__builtin_amdgcn_s_wait_asynccnt)
#define WAIT_ASYNC(n) __builtin_amdgcn_s_wait_asynccnt(n)
#else
#define WAIT_ASYNC(n) asm volatile("s_wait_asynccnt %0" ::"i"(n) : "memory")
#endif

__global__ __launch_bounds__(R) void ealif_scan_kernel(
    const float* __restrict__ x, const float* __restrict__ Vth,
    float* __restrict__ out, int C, int T, int nrows)
{
    __shared__ float tile[2][R * STRIDE];

    const int tid = threadIdx.x;
    const int r0  = blockIdx.x * R;
    const int r   = r0 + tid;
    const float vth = Vth[(r < nrows ? r : 0) % C];
    const int NC = T / TC;

    float mem = 0.f, spk = 0.f, tha = 0.f;

#if HAVE_ASYNC_LDS
    auto fill_async = [&](int k, int b) {
#pragma unroll
        for (int i = 0; i < F4ROW; ++i) {
            int v   = i * R + tid;
            int row = v / F4ROW;        // 8 consecutive lanes cover one row: 128B coalesced
            int tq  = v % F4ROW;
            int gr  = r0 + row;
            if (gr >= nrows) gr = nrows - 1;   // clamp (LDS dests still unique)
            const float* gp = x + (size_t)gr * T + (size_t)k * TC + tq * 4;
            float*       lp = &tile[b][row * STRIDE + tq * 4];
            __builtin_amdgcn_global_load_async_to_lds_b128(
                (__attribute__((address_space(1))) v4i*)gp,
                (__attribute__((address_space(3))) v4i*)lp,
                /*offset=*/0, /*cpol=*/0);
        }
    };
    fill_async(0, 0);                   // prologue: chunk 0 in flight
#else
    auto fill_sync = [&](int k, int b) {
#pragma unroll
        for (int i = 0; i < F4ROW; ++i) {
            int v   = i * R + tid;
            int row = v / F4ROW;
            int tq  = v % F4ROW;
            int gr  = r0 + row;
            if (gr >= nrows) gr = nrows - 1;
            f4 d = __builtin_nontemporal_load(
                (const f4*)(x + (size_t)gr * T + (size_t)k * TC + tq * 4));
            *(f4*)&tile[b][row * STRIDE + tq * 4] = d;
        }
    };
#endif

    for (int k = 0; k < NC; ++k) {
        const int b = k & 1;

#if HAVE_ASYNC_LDS
        // Issue next chunk's async loads, then wait for current chunk.
        // Only loads are outstanding on ASYNCcnt and loads complete in order,
        // so "<= LPT" guarantees chunk k has fully landed in LDS.
        if (k + 1 < NC) { fill_async(k + 1, (k + 1) & 1); WAIT_ASYNC(LPT); }
        else            { WAIT_ASYNC(0); }
#else
        fill_sync(k, b);
#endif
        __syncthreads();                 // per-wave wait + block barrier: tile[b] valid

        // --- sequential eALIF scan over this time chunk (row in LDS) ---
        float* rowp = &tile[b][tid * STRIDE];
#pragma unroll
        for (int t = 0; t < TC; ++t) {
            float xi = rowp[t];
            mem = (mem - spk * vth) * EALIF_BETA + xi;   // leaky integrate + reset
            tha = EALIF_RHO * tha + spk;                 // threshold adaptation
            float ath = vth + EALIF_GAMMA * tha;         // adaptive threshold
            spk = (mem - ath) > 0.f ? 1.f : 0.f;         // Heaviside spike
            rowp[t] = spk;                               // overwrite x with spike in place
        }
        __syncthreads();                 // all rows of tile[b] now hold spikes

        // --- coalesced, non-temporal streaming store of the spike tile ---
#pragma unroll
        for (int i = 0; i < F4ROW; ++i) {
            int v   = i * R + tid;
            int row = v / F4ROW;
            int tq  = v % F4ROW;
            int gr  = r0 + row;
            if (gr < nrows) {
                f4 d = *(const f4*)&tile[b][row * STRIDE + tq * 4];
                __builtin_nontemporal_store(
                    d, (f4*)(out + (size_t)gr * T + (size_t)k * TC + tq * 4));
            }
        }
        __syncthreads();                 // release tile[b] before it is refilled (chunk k+2)
    }
}

extern "C" void kernel_launch(void* const* d_in, const int* in_sizes, int n_in,
                              void* d_out, int out_size, void* d_ws, size_t ws_size,
                              hipStream_t stream) {
    const float* x   = (const float*)d_in[0];   // [B, C, T] fp32
    const float* vth = (const float*)d_in[1];   // [C] fp32
    float*       out = (float*)d_out;           // [B, C, T] fp32 spikes

    const int C = in_sizes[1];                  // 1024
    const int T = 512;                          // fixed by reference
    const int nrows = in_sizes[0] / T;          // B*C = 65536 independent rows
    const int grid  = (nrows + R - 1) / R;      // 512 blocks of 4 waves

    ealif_scan_kernel<<<grid, R, 0, stream>>>(x, vth, out, C, T, nrows);
}